// CompGCNBase_86260123173503
// MI455X (gfx1250) — compile-verified
//
#include <hip/hip_runtime.h>
#include <hip/hip_bf16.h>
#include <math.h>

typedef __attribute__((ext_vector_type(2))) float v2f;
typedef __attribute__((ext_vector_type(8))) float v8f;

#define AS1 __attribute__((address_space(1)))
typedef AS1 const float* gcfp;   // global const float*
typedef AS1 float*       gfp;    // global float*
typedef AS1 const v2f*   gcv2p;  // global const float2*

__device__ __forceinline__ gcfp to_gc(const float* p) { return (gcfp)(unsigned long long)p; }
__device__ __forceinline__ gfp  to_g (float* p)       { return (gfp) (unsigned long long)p; }

#define DIM 100
#define BN_EPS 1e-5f

// ---------------------------------------------------------------- degree ----
__global__ void k_degree(const int* __restrict__ rows, float* __restrict__ deg, int E) {
    int e = blockIdx.x * blockDim.x + threadIdx.x;
    if (e < E) atomicAdd(&deg[rows[e]], 1.0f);
}

__global__ void k_deginv(float* __restrict__ deg, int n) {
    int i = blockIdx.x * blockDim.x + threadIdx.x;
    if (i < n) {
        float d = deg[i];
        deg[i] = (d > 0.f) ? rsqrtf(fmaxf(d, 1.f)) : 0.f;
    }
}

// ------------------------------------------------- edge gather/scatter ------
// acc[row, :] += dinv[row]*dinv[col] * x[col, :] * rel[etype, :]
__global__ void k_edge_accum(const int* __restrict__ rows, const int* __restrict__ cols,
                             const int* __restrict__ types,
                             const float* __restrict__ x, const float* __restrict__ rel,
                             const float* __restrict__ dinv, float* __restrict__ acc, int E) {
    int e = blockIdx.x;
    int d = threadIdx.x;
    if (e >= E || d >= DIM) return;
    int r = rows[e];
    int c = cols[e];
    int t = types[e];
    float norm = dinv[r] * dinv[c];
    float v = x[(long long)c * DIM + d] * rel[(long long)t * DIM + d] * norm;
    atomicAdd(&acc[(long long)r * DIM + d], v);
}

// --------------------------------------------------------- w_loop scale -----
__global__ void k_scale_wloop(const float* __restrict__ w_loop,
                              const float* __restrict__ loop_rel, float* __restrict__ wl) {
    int i = blockIdx.x * blockDim.x + threadIdx.x;
    if (i < DIM * DIM) wl[i] = loop_rel[i / DIM] * w_loop[i];
}

// --------------------------------------------------------- WMMA GEMM --------
// C = alpha * (sum_m A[m] @ B[m]) + bias.  All f32, row-major, N = K = DIM.
// One wave per 16x16 output tile; K stepped by 4 via V_WMMA_F32_16X16X4_F32.
// M must be a multiple of 16. N masking is done by COLUMN CLAMPING: B column
// n only contributes to output column n, so out-of-range lanes may compute
// garbage that is simply never stored — no per-iteration EXEC manipulation.
template <int NMAT>
__global__ void k_gemm_wmma(const float* A0_, const float* B0_,
                            const float* A1_, const float* B1_,
                            const float* A2_, const float* B2_,
                            int M, float alpha, const float* bias_, float* C_) {
    const int lane = threadIdx.x & 31;
    const int wave = threadIdx.x >> 5;
    const int wavesPerBlock = blockDim.x >> 5;
    const int ntn = (DIM + 15) >> 4;                 // 7
    const int ntm = M >> 4;
    const int tile = blockIdx.x * wavesPerBlock + wave;
    if (tile >= ntm * ntn) return;                   // wave-uniform: EXEC stays full

    const int tm = tile / ntn;
    const int tn = tile % ntn;
    const int mbase = tm << 4;
    const int nbase = tn << 4;
    const int half = lane >> 4;                      // 0: K=k,k+1 / M=v ; 1: K=k+2,k+3 / M=v+8
    const int l = lane & 15;
    const int col = nbase + l;
    const int colc = (col < DIM) ? col : (DIM - 1);  // clamp instead of mask

    gcfp A[NMAT];
    gcfp B[NMAT];
    A[0] = to_gc(A0_); B[0] = to_gc(B0_);
    if constexpr (NMAT > 1) { A[1] = to_gc(A1_); B[1] = to_gc(B1_); }
    if constexpr (NMAT > 2) { A[2] = to_gc(A2_); B[2] = to_gc(B2_); }

    v8f c = {0.f, 0.f, 0.f, 0.f, 0.f, 0.f, 0.f, 0.f};

    const int arow = (mbase + l) * DIM + 2 * half;   // per-lane A row base
    const int kh = 2 * half;

#pragma unroll
    for (int m = 0; m < NMAT; ++m) {
        gcfp Am = A[m];
        gcfp Bm = B[m];
#pragma unroll
        for (int k = 0; k < DIM; k += 4) {
            // A 16x4: lanes 0-15 carry K=k,k+1 ; lanes 16-31 carry K=k+2,k+3 (contiguous b64)
            v2f a = *(gcv2p)(Am + arow + k);
            // B 4x16: same half-wave K split; unconditional clamped loads
            v2f b;
            b.x = Bm[(k + kh) * DIM + colc];
            b.y = Bm[(k + kh + 1) * DIM + colc];
            c = __builtin_amdgcn_wmma_f32_16x16x4_f32(false, a, false, b, (short)0, c,
                                                      false, false);
        }
    }

    if (col < DIM) {
        float bv = bias_ ? bias_[col] : 0.f;
        gfp C = to_g(C_);
#pragma unroll
        for (int v = 0; v < 8; ++v) {
            // C/D layout: VGPR v -> M=v (lanes 0-15), M=v+8 (lanes 16-31)
            C[(mbase + v + 8 * half) * DIM + col] = c[v] * alpha + bv;
        }
    }
}

// ------------------------------------------------------------ BN stats ------
__global__ void k_bn_stats(const float* __restrict__ P, float* __restrict__ sum,
                           float* __restrict__ sumsq, int total) {
    __shared__ float s[DIM];
    __shared__ float sq[DIM];
    for (int i = threadIdx.x; i < DIM; i += blockDim.x) { s[i] = 0.f; sq[i] = 0.f; }
    __syncthreads();
    for (int i = blockIdx.x * blockDim.x + threadIdx.x; i < total;
         i += gridDim.x * blockDim.x) {
        float v = P[i];
        int c = i % DIM;
        atomicAdd(&s[c], v);
        atomicAdd(&sq[c], v * v);
    }
    __syncthreads();
    for (int i = threadIdx.x; i < DIM; i += blockDim.x) {
        atomicAdd(&sum[i], s[i]);
        atomicAdd(&sumsq[i], sq[i]);
    }
}

__global__ void k_bn_final(const float* __restrict__ sum, const float* __restrict__ sumsq,
                           const float* __restrict__ gamma, const float* __restrict__ beta,
                           float* __restrict__ scale, float* __restrict__ shift, int M) {
    int c = blockIdx.x * blockDim.x + threadIdx.x;
    if (c < DIM) {
        float inv = 1.f / (float)M;
        float mu = sum[c] * inv;
        float var = sumsq[c] * inv - mu * mu;
        float sc = gamma[c] * rsqrtf(var + BN_EPS);
        scale[c] = sc;
        shift[c] = beta[c] - mu * sc;
    }
}

__global__ void k_apply(const float* __restrict__ P, const float* __restrict__ scale,
                        const float* __restrict__ shift, float* __restrict__ out, int total) {
    for (int i = blockIdx.x * blockDim.x + threadIdx.x; i < total;
         i += gridDim.x * blockDim.x) {
        int c = i % DIM;
        out[i] = tanhf(P[i] * scale[c] + shift[c]);
    }
}

// ------------------------------------------------------------- launcher -----
extern "C" void kernel_launch(void* const* d_in, const int* in_sizes, int n_in,
                              void* d_out, int out_size, void* d_ws, size_t ws_size,
                              hipStream_t stream) {
    const float* x        = (const float*)d_in[0];
    const int*   ei       = (const int*)d_in[1];   // (2, 2E) flat
    const int*   et       = (const int*)d_in[2];   // (2E,)
    const float* rel      = (const float*)d_in[3]; // (400, 100)
    const float* w_in     = (const float*)d_in[4];
    const float* w_out    = (const float*)d_in[5];
    const float* w_loop   = (const float*)d_in[6];
    const float* w_rel    = (const float*)d_in[7];
    const float* loop_rel = (const float*)d_in[8];
    const float* bias     = (const float*)d_in[9];
    const float* gamma    = (const float*)d_in[10];
    const float* beta     = (const float*)d_in[11];

    const int nent = in_sizes[0] / DIM;      // 50000
    const int E2   = in_sizes[2];            // 2E = 1,000,000
    const int E    = E2 / 2;                 // 500,000
    const int nrel = in_sizes[3] / DIM;      // 400

    // Workspace (floats): [acc_in | acc_out | deg_in | deg_out | sum | sumsq | P | wl | scale | shift]
    float* ws      = (float*)d_ws;
    float* acc_in  = ws;
    float* acc_out = acc_in  + (long long)nent * DIM;
    float* deg_in  = acc_out + (long long)nent * DIM;
    float* deg_out = deg_in  + nent;
    float* sum     = deg_out + nent;
    float* sumsq   = sum     + DIM;
    float* P       = sumsq   + DIM;
    float* wl      = P       + (long long)nent * DIM;
    float* scale   = wl      + DIM * DIM;
    float* shift   = scale   + DIM;

    size_t zbytes = (size_t)((char*)P - (char*)ws);
    hipMemsetAsync(ws, 0, zbytes, stream);

    const int tpb = 256;

    // Degrees per direction (reference computes deg from 'row' only)
    k_degree<<<(E + tpb - 1) / tpb, tpb, 0, stream>>>(ei,     deg_in,  E);
    k_degree<<<(E + tpb - 1) / tpb, tpb, 0, stream>>>(ei + E, deg_out, E);
    k_deginv<<<(nent + tpb - 1) / tpb, tpb, 0, stream>>>(deg_in,  nent);
    k_deginv<<<(nent + tpb - 1) / tpb, tpb, 0, stream>>>(deg_out, nent);

    // Edge accumulation (GEMM hoisted out of the edge loop by linearity)
    k_edge_accum<<<E, 128, 0, stream>>>(ei,     ei + 2 * E, et,     x, rel, deg_in,  acc_in,  E);
    k_edge_accum<<<E, 128, 0, stream>>>(ei + E, ei + 3 * E, et + E, x, rel, deg_out, acc_out, E);

    // Fold loop_rel into w_loop
    k_scale_wloop<<<(DIM * DIM + tpb - 1) / tpb, tpb, 0, stream>>>(w_loop, loop_rel, wl);

    // Fused 3-way WMMA GEMM: P = (acc_in@w_in + acc_out@w_out + x@wl)/3 + bias
    {
        const int ntn = (DIM + 15) / 16;                 // 7
        const int tiles = (nent / 16) * ntn;             // 21875
        const int wpb = 4;                               // 4 wave32 / block
        k_gemm_wmma<3><<<(tiles + wpb - 1) / wpb, 32 * wpb, 0, stream>>>(
            acc_in, w_in, acc_out, w_out, x, wl, nent, 1.f / 3.f, bias, P);
    }

    // BatchNorm statistics + finalize + tanh apply
    const int total = nent * DIM;
    k_bn_stats<<<512, tpb, 0, stream>>>(P, sum, sumsq, total);
    k_bn_final<<<1, 128, 0, stream>>>(sum, sumsq, gamma, beta, scale, shift, nent);
    k_apply<<<2048, tpb, 0, stream>>>(P, scale, shift, (float*)d_out, total);

    // Second output: rel_embed @ w_rel  (== (rel_all @ w_rel)[:-1])
    {
        const int ntn = (DIM + 15) / 16;
        const int tiles = (nrel / 16) * ntn;             // 175
        const int wpb = 4;
        k_gemm_wmma<1><<<(tiles + wpb - 1) / wpb, 32 * wpb, 0, stream>>>(
            rel, w_rel, nullptr, nullptr, nullptr, nullptr, nrel,
            1.f, nullptr, (float*)d_out + (long long)nent * DIM);
    }

    (void)n_in; (void)out_size; (void)ws_size;
}